// FixedPairSelector_86277303042728
// MI455X (gfx1250) — compile-verified
//
#include <hip/hip_runtime.h>

// Reference: BATCH=16384 rows, B=1024 cols, S=2048 index pairs.
// out[b, s, 0] = xB[b, idxL[s]];  out[b, s, 1] = xB[b, idxR[s]]
// Pure gather: store-bandwidth bound (~256 MiB out @ 23.3 TB/s ≈ 11-14 us floor).

#define BATCH 16384
#define BCOLS 1024
#define SIDX  2048
#define TPB   256

// Native clang vector types (HIP float4/int2 are classes and are rejected by
// __builtin_nontemporal_store).
typedef float v4f __attribute__((ext_vector_type(4)));
typedef int   v2i __attribute__((ext_vector_type(2)));

__global__ __launch_bounds__(TPB)
void FixedPairSelector_gather_kernel(const float* __restrict__ xB,
                                     const int*   __restrict__ idxL,
                                     const int*   __restrict__ idxR,
                                     float*       __restrict__ out)
{
    __shared__ float row[BCOLS];            // 4 KiB of the 320 KiB/WGP LDS

    const int b = blockIdx.x;               // one batch row per block
    const int t = threadIdx.x;

    // Prefetch the index arrays (16 KiB total) while we stage the row.
    // 256 threads x 8 entries covers all 2048 entries of each array.
    // Emits global_prefetch_b8 on gfx1250.
    __builtin_prefetch(idxL + t * 8, 0, 3);
    __builtin_prefetch(idxR + t * 8, 0, 3);

    // Stage this row of xB into LDS: 1024 floats / 256 threads = one 16B
    // (global_load_b128 -> ds_store_b128) per thread, fully coalesced.
    const v4f* rowg = reinterpret_cast<const v4f*>(xB + (size_t)b * BCOLS);
    reinterpret_cast<v4f*>(row)[t] = rowg[t];
    __syncthreads();

    // Output base for this row: 2*S floats.
    float* outb = out + (size_t)b * (2 * SIDX);

    // Each thread emits 4 x 16B stores covering 8 s-values total.
    // Lane-consecutive 16B stores -> perfectly coalesced 512B per wave op.
    #pragma unroll
    for (int j = 0; j < 4; ++j) {
        const int s2 = j * (2 * TPB) + t * 2;                      // even s base
        const v2i l = *reinterpret_cast<const v2i*>(idxL + s2);    // 8B coalesced
        const v2i r = *reinterpret_cast<const v2i*>(idxR + s2);

        v4f o;                      // [ L(s2), R(s2), L(s2+1), R(s2+1) ]
        o.x = row[l.x];             // ds_load_b32 gathers from LDS
        o.y = row[r.x];
        o.z = row[l.y];
        o.w = row[r.y];

        // Write-once streaming output: non-temporal (TH.NT) b128 store so the
        // 256 MiB of output does not evict xB/idx working set from L2.
        __builtin_nontemporal_store(o, reinterpret_cast<v4f*>(outb + s2 * 2));
    }
}

extern "C" void kernel_launch(void* const* d_in, const int* in_sizes, int n_in,
                              void* d_out, int out_size, void* d_ws, size_t ws_size,
                              hipStream_t stream) {
    const float* xB   = (const float*)d_in[0];
    const int*   idxL = (const int*)d_in[1];
    const int*   idxR = (const int*)d_in[2];
    float*       out  = (float*)d_out;

    dim3 grid(BATCH);   // 16384 blocks, one batch row each
    dim3 block(TPB);    // 256 threads = 8 wave32s
    FixedPairSelector_gather_kernel<<<grid, block, 0, stream>>>(xB, idxL, idxR, out);
}